// BernsteinNetwork_70093866270849
// MI455X (gfx1250) — compile-verified
//
#include <hip/hip_runtime.h>
#include <hip/hip_bf16.h>
#include <math.h>

typedef __attribute__((ext_vector_type(2))) float v2f;
typedef __attribute__((ext_vector_type(8))) float v8f;

#define NPTS  32768
#define DDIM  8
#define OP1   25          // ORDER+1
#define PPERM 32
#define KPAD  28
#define MATSZ (KPAD*KPAD) // 784 floats per (padded) 28x28 matrix
#define NDROW (DDIM*OP1)  // 200: ball floats per point row

// ---- workspace layout (float offsets) ----
#define WS_SC2   0                      // 25 (pad 32)
#define WS_WV0   32                     // 32*25 = 800
#define WS_WVR   832                    // 7*32*625 = 140000
#define WS_BALL  140832                 // 32768*8*25 = 6553600
#define WS_PART  6694432                // 32*32768*2 = 2097152
// total = 8791584 floats (~35.2 MB)

__device__ __constant__ float c_binom[25] = {
  1.f, 24.f, 276.f, 2024.f, 10626.f, 42504.f, 134596.f, 346104.f,
  735471.f, 1307504.f, 1961256.f, 2496144.f, 2704156.f, 2496144.f,
  1961256.f, 1307504.f, 735471.f, 346104.f, 134596.f, 42504.f,
  10626.f, 2024.f, 276.f, 24.f, 1.f
};

// ---- K1: sc2[k] = rowsum(inv(B(grid)^2)) via double Gauss-Jordan (1 thread) ----
__global__ void k_prior_scale(float* ws) {
  if (threadIdx.x != 0) return;
  double M[25][25], Inv[25][25];
  for (int i = 0; i < 25; ++i) {
    double x = (double)i / 24.0;
    double om = 1.0 - x;
    double fwd[25]; double f = 1.0;
    for (int k = 0; k < 25; ++k) { fwd[k] = f; f *= x; }
    double b = 1.0;
    for (int k = 24; k >= 0; --k) {
      double v = fwd[k] * b * (double)c_binom[k];
      M[i][k] = v * v;
      b *= om;
    }
  }
  for (int i = 0; i < 25; ++i)
    for (int j = 0; j < 25; ++j) Inv[i][j] = (i == j) ? 1.0 : 0.0;
  for (int c = 0; c < 25; ++c) {
    int piv = c; double best = fabs(M[c][c]);
    for (int r = c + 1; r < 25; ++r) {
      double a = fabs(M[r][c]);
      if (a > best) { best = a; piv = r; }
    }
    if (piv != c) {
      for (int j = 0; j < 25; ++j) {
        double t = M[c][j];  M[c][j]  = M[piv][j];  M[piv][j]  = t;
        t = Inv[c][j]; Inv[c][j] = Inv[piv][j]; Inv[piv][j] = t;
      }
    }
    double d = 1.0 / M[c][c];
    for (int j = 0; j < 25; ++j) { M[c][j] *= d; Inv[c][j] *= d; }
    for (int r = 0; r < 25; ++r) {
      if (r == c) continue;
      double fr = M[r][c];
      if (fr != 0.0)
        for (int j = 0; j < 25; ++j) { M[r][j] -= fr * M[c][j]; Inv[r][j] -= fr * Inv[c][j]; }
    }
  }
  for (int k = 0; k < 25; ++k) {
    double s = 0.0;
    for (int j = 0; j < 25; ++j) s += Inv[k][j];
    ws[WS_SC2 + k] = (float)s;
  }
}

// ---- K2: variance weight pre-transform: exp(w)*sc2[col] ----
__global__ void k_prep_var(const float* __restrict__ varw0,
                           const float* __restrict__ varw_rest,
                           float* __restrict__ ws) {
  int idx = blockIdx.x * blockDim.x + threadIdx.x;
  const int n0 = PPERM * OP1;                 // 800
  const int total = n0 + 7 * PPERM * OP1 * OP1;
  if (idx >= total) return;
  if (idx < n0) {
    int k = idx % OP1;
    ws[WS_WV0 + idx] = expf(varw0[idx]) * ws[WS_SC2 + k];
  } else {
    int e = idx - n0;
    int j = e % OP1;
    ws[WS_WVR + e] = expf(varw_rest[e]) * ws[WS_SC2 + j];
  }
}

// ---- K3: Bernstein basis table Ball[n][d][k] ----
__global__ void k_bernstein(const float* __restrict__ Xnew, float* __restrict__ ws) {
  int idx = blockIdx.x * blockDim.x + threadIdx.x;   // n*8 + d
  if (idx >= NPTS * DDIM) return;
  float x = Xnew[idx];
  float om = 1.f - x;
  float fwd[OP1]; float f = 1.f;
  #pragma unroll
  for (int k = 0; k < OP1; ++k) { fwd[k] = f; f *= x; }
  float b = 1.f;
  float* out = ws + WS_BALL + (size_t)idx * OP1;
  #pragma unroll
  for (int k = OP1 - 1; k >= 0; --k) { out[k] = fwd[k] * b * c_binom[k]; b *= om; }
}

// ---- K4: WMMA chain kernel ----
// LDS layout (floats). Weights stored PAIR-INTERLEAVED:
//   W[step][kpair q][col j] pair = { W[2q][j], W[2q+1][j] } at  step*784 + q*56 + 2*j (+0/1)
#define LDS_WM 0
#define LDS_WV (7*MATSZ)                       // 5488
#define LDS_ST (2*7*MATSZ)                     // 10976
#define LDS_W0 (LDS_ST + 4*2*(16*KPAD))        // 10976 + 3584 = 14560
#define LDS_TOTAL (LDS_W0 + 64 + 32)           // 14656 floats = 58624 B

__global__ __launch_bounds__(128)
void k_chain(const int*   __restrict__ perm,
             const float* __restrict__ meanw0,
             const float* __restrict__ meanw_rest,
             const float* __restrict__ post_prec,
             float*       __restrict__ ws) {
  __shared__ float sm[LDS_TOTAL];
  const int tid  = threadIdx.x;
  const int lane = tid & 31;
  const int wave = tid >> 5;
  const int p = blockIdx.y;
  const int wg_row0 = blockIdx.x * 256;

  // zero-init all LDS (guarantees 0.0 in K=25..27 padding rows/cols)
  for (int i = tid; i < LDS_TOTAL; i += 128) sm[i] = 0.f;
  __syncthreads();

  // stage this permutation's 7 mean + 7 var matrices, pair-interleaved, padded 25->28
  for (int i = tid; i < 7 * OP1 * OP1; i += 128) {
    int s = i / (OP1 * OP1);
    int rem = i - s * (OP1 * OP1);
    int k = rem / OP1, j = rem - k * OP1;
    int dst = s * MATSZ + (k >> 1) * (2 * KPAD) + 2 * j + (k & 1);
    sm[LDS_WM + dst] = meanw_rest[(s*PPERM + p)*OP1*OP1 + rem];
    sm[LDS_WV + dst] = ws[WS_WVR + (size_t)(s*PPERM + p)*OP1*OP1 + rem];
  }
  // stage step-0 weight vectors
  if (tid < OP1) {
    sm[LDS_W0 + tid]      = meanw0[p*OP1 + tid];
    sm[LDS_W0 + 32 + tid] = ws[WS_WV0 + p*OP1 + tid];
  }
  __syncthreads();

  const float* ball = ws + WS_BALL;
  float* stm = sm + LDS_ST + wave * (2 * 16 * KPAD);
  float* stv = stm + 16 * KPAD;
  const int lhalf = (lane < 16) ? 1 : 0;
  const int arow  = lhalf ? lane : (lane - 16);   // M row in 16-row tile
  const int kofs  = lhalf ? 0 : 2;                // A/B VGPR0 K offset
  const int c0    = lane & 15;
  const float inv_pp = 1.f / post_prec[p];
  const int d0 = perm[p * DDIM + 0];
  const float* b0p = ball + d0 * OP1;             // uniform base for step 0

  for (int t = 0; t < 4; ++t) {
    const int row0 = wg_row0 + wave * 64 + t * 16;

    // 32-bit per-lane row offsets for Bernstein gathers (C-layout rows)
    unsigned rbase[8];
    #pragma unroll
    for (int r = 0; r < 8; ++r) {
      int row = lhalf ? r : (r + 8);
      rbase[r] = (unsigned)((row0 + row) * NDROW);
    }

    // step i=0: fm = w0 .* B0 ; fv = (exp(wv0)*sc2) .* B0^2
    for (int e = lane; e < 16 * OP1; e += 32) {
      int r = e / OP1, k = e - r * OP1;
      float bi = b0p[(unsigned)((row0 + r) * NDROW) + (unsigned)k];
      stm[r*KPAD + k] = sm[LDS_W0 + k] * bi;
      stv[r*KPAD + k] = sm[LDS_W0 + 32 + k] * bi * bi;
    }
    __builtin_amdgcn_wave_barrier();

    #pragma unroll
    for (int i = 1; i < DDIM; ++i) {
      const int di = perm[p * DDIM + i];             // SGPR (uniform)
      const float* bp = ball + di * OP1;             // uniform base
      const float* wm = sm + LDS_WM + (i - 1) * MATSZ;
      const float* wv = sm + LDS_WV + (i - 1) * MATSZ;
      v8f am0 = {}, am1 = {}, av0 = {}, av1 = {};
      #pragma unroll
      for (int kk = 0; kk < KPAD; kk += 4) {
        const int ka = kk + kofs;                    // even
        const int q  = ka >> 1;                      // K row-pair
        // A fragments: contiguous aligned pair in staging
        v2f amf = *(const v2f*)(stm + arow*KPAD + ka);
        v2f avf = *(const v2f*)(stv + arow*KPAD + ka);
        // B fragments: contiguous aligned pair in pair-interleaved weights
        v2f bm0 = *(const v2f*)(wm + q*(2*KPAD) + 2*c0);
        v2f bm1 = *(const v2f*)(wm + q*(2*KPAD) + 2*(16 + c0));
        v2f bv0 = *(const v2f*)(wv + q*(2*KPAD) + 2*c0);
        v2f bv1 = *(const v2f*)(wv + q*(2*KPAD) + 2*(16 + c0));
        am0 = __builtin_amdgcn_wmma_f32_16x16x4_f32(false, amf, false, bm0, (short)0, am0, false, false);
        am1 = __builtin_amdgcn_wmma_f32_16x16x4_f32(false, amf, false, bm1, (short)0, am1, false, false);
        av0 = __builtin_amdgcn_wmma_f32_16x16x4_f32(false, avf, false, bv0, (short)0, av0, false, false);
        av1 = __builtin_amdgcn_wmma_f32_16x16x4_f32(false, avf, false, bv1, (short)0, av1, false, false);
      }
      __builtin_amdgcn_wave_barrier();   // all lanes done reading old staging (wave-private)
      // D-tile (C layout): VGPR r -> rows r (lanes 0-15) / r+8 (lanes 16-31), col = ct*16 + lane%16
      #pragma unroll
      for (int ct = 0; ct < 2; ++ct) {
        int col = ct * 16 + c0;
        if (col < OP1) {
          #pragma unroll
          for (int r = 0; r < 8; ++r) {
            int row = lhalf ? r : (r + 8);
            float bi = bp[rbase[r] + (unsigned)col];
            float m = (ct == 0) ? am0[r] : am1[r];
            float v = (ct == 0) ? av0[r] : av1[r];
            stm[row*KPAD + col] = m * bi;
            stv[row*KPAD + col] = v * bi * bi;
          }
        }
      }
      __builtin_amdgcn_wave_barrier();
    }

    // per-row reduction -> deterministic per-perm partials
    float s = 0.f;
    if (lhalf) {
      #pragma unroll
      for (int j = 0; j < OP1; ++j) s += stm[arow*KPAD + j];
      ws[WS_PART + (size_t)(p * NPTS + row0 + arow) * 2 + 0] = s;
    } else {
      #pragma unroll
      for (int j = 0; j < OP1; ++j) s += stv[arow*KPAD + j];
      ws[WS_PART + (size_t)(p * NPTS + row0 + arow) * 2 + 1] = s * inv_pp;
    }
    __builtin_amdgcn_wave_barrier();
  }
}

// ---- K5: fixed-order reduce over permutations ----
__global__ void k_reduce(const float* __restrict__ ws, float* __restrict__ out) {
  int n = blockIdx.x * blockDim.x + threadIdx.x;
  if (n >= NPTS) return;
  float m = 0.f, v = 0.f;
  for (int p = 0; p < PPERM; ++p) {
    m += ws[WS_PART + (size_t)(p * NPTS + n) * 2 + 0];
    v += ws[WS_PART + (size_t)(p * NPTS + n) * 2 + 1];
  }
  out[n * 2 + 0] = m;
  out[n * 2 + 1] = v;
}

extern "C" void kernel_launch(void* const* d_in, const int* in_sizes, int n_in,
                              void* d_out, int out_size, void* d_ws, size_t ws_size,
                              hipStream_t stream) {
  const float* Xnew       = (const float*)d_in[0];
  const int*   perm       = (const int*)  d_in[1];
  const float* meanw0     = (const float*)d_in[2];
  const float* meanw_rest = (const float*)d_in[3];
  const float* varw0      = (const float*)d_in[4];
  const float* varw_rest  = (const float*)d_in[5];
  const float* post_prec  = (const float*)d_in[6];
  float* ws  = (float*)d_ws;
  float* out = (float*)d_out;

  k_prior_scale<<<1, 32, 0, stream>>>(ws);
  {
    int total = PPERM * OP1 + 7 * PPERM * OP1 * OP1;
    k_prep_var<<<(total + 255) / 256, 256, 0, stream>>>(varw0, varw_rest, ws);
  }
  k_bernstein<<<(NPTS * DDIM + 255) / 256, 256, 0, stream>>>(Xnew, ws);
  {
    dim3 grid(NPTS / 256, PPERM);
    k_chain<<<grid, 128, 0, stream>>>(perm, meanw0, meanw_rest, post_prec, ws);
  }
  k_reduce<<<(NPTS + 255) / 256, 256, 0, stream>>>(ws, out);
}